// EncoderSRN_73993696575783
// MI455X (gfx1250) — compile-verified
//
#include <hip/hip_runtime.h>
#include <hip/hip_bf16.h>

// EncoderSRN: x_proj = emb[tokens] @ wx_w.T + wx_b  (WMMA f32 GEMM)
//             then sequential h = sigmoid(xp_t + h @ wh_w.T + wh_b), out = h_final
#define VOCAB 128000
#define EMB 256
#define HID 256
#define SEQ 4096

typedef __attribute__((ext_vector_type(2))) float v2f;
typedef __attribute__((ext_vector_type(4))) float v4f;
typedef __attribute__((ext_vector_type(8))) float v8f;
typedef __attribute__((ext_vector_type(4))) int   v4i;

#define A_STRIDE 260  // 16B-aligned row stride, 260 % 64 == 4 -> conflict-free tile reads

// gfx1250 async global->LDS copy (ASYNCcnt-tracked), guarded so compile never breaks.
#if defined(__has_builtin)
#if __has_builtin(__builtin_amdgcn_global_load_async_to_lds_b128)
#define HAVE_ASYNC_LDS 1
#endif
#endif
#ifndef HAVE_ASYNC_LDS
#define HAVE_ASYNC_LDS 0
#endif

typedef v4i __attribute__((address_space(1)))* g_v4ip;   // global AS pointer
typedef v4i __attribute__((address_space(3)))* l_v4ip;   // LDS AS pointer

// ---------------------------------------------------------------------------
// Phase 1: x_proj[m][n] = sum_k emb[tokens[m]][k] * wx_w[n][k] + wx_b[n]
// Grid: SEQ/16 blocks x 256 threads (8 waves). Each block: one 16-row M tile.
// Each wave: two 16-col N tiles, K-loop of V_WMMA_F32_16X16X4_F32.
// A tile gathered into LDS via async-LDS loads when available.
// ---------------------------------------------------------------------------
__global__ __launch_bounds__(256) void xproj_wmma_kernel(
    const int* __restrict__ tokens,
    const float* __restrict__ embedding,
    const float* __restrict__ wx_w,
    const float* __restrict__ wx_b,
    float* __restrict__ xp)
{
    __shared__ float As[16 * A_STRIDE];   // 16 x 256 fp32 A tile (gathered embeddings)

    const int m0 = blockIdx.x * 16;
    const int t  = threadIdx.x;

    // Gather A tile: thread t handles 4 float4s of row (t>>4).
    {
        const int row = t >> 4;                          // 0..15
        const int tok = tokens[m0 + row];
        const float* src = embedding + (long)tok * EMB;
        const int c4 = t & 15;                           // float4 column group
#if HAVE_ASYNC_LDS
        #pragma unroll
        for (int i = 0; i < 4; ++i) {
            const int col = (c4 + i * 16) * 4;           // 0..252, 16B aligned
            // per-lane gather: global b128 -> LDS, tracked by ASYNCcnt
            __builtin_amdgcn_global_load_async_to_lds_b128(
                (g_v4ip)(src + col),
                (l_v4ip)(&As[row * A_STRIDE + col]),
                0, 0);
        }
        asm volatile("s_wait_asynccnt 0x0" ::: "memory");
#else
        #pragma unroll
        for (int i = 0; i < 4; ++i) {
            const int col = (c4 + i * 16) * 4;
            v4f v = *(const v4f*)(src + col);
            *(v4f*)(&As[row * A_STRIDE + col]) = v;
        }
#endif
    }
    __syncthreads();

    const int lane = t & 31;
    const int wave = t >> 5;                 // 0..7
    const int m    = lane & 15;              // A row this lane feeds
    const int kp   = (lane >> 4) * 2;        // K sub-offset: lanes 0-15 -> K0,K1; 16-31 -> K2,K3

    #pragma unroll
    for (int tile = 0; tile < 2; ++tile) {
        const int n = (wave * 2 + tile) * 16 + (lane & 15);   // output column
        const float* brow = wx_w + (long)n * EMB;             // B[k][n] = wx_w[n][k]

        v8f c = {};
        #pragma unroll 4
        for (int k0 = 0; k0 < EMB; k0 += 4) {
            // A 16x4 layout: lane holds A[m][k0+kp], A[m][k0+kp+1]
            v2f a = *(const v2f*)(&As[m * A_STRIDE + k0 + kp]);
            // B 4x16 layout: lane holds B[k0+kp][n], B[k0+kp+1][n] (contiguous in wx_w row n)
            v2f b = *(const v2f*)(brow + k0 + kp);
            c = __builtin_amdgcn_wmma_f32_16x16x4_f32(
                    /*neg_a=*/false, a, /*neg_b=*/false, b,
                    /*c_mod=*/(short)0, c, /*reuse_a=*/false, /*reuse_b=*/false);
        }

        // C/D layout: VGPR r -> row r (lanes 0-15) or row 8+r (lanes 16-31), col = lane&15
        const float bias  = wx_b[n];
        const int   rbase = (lane >> 4) * 8;
        #pragma unroll
        for (int r = 0; r < 8; ++r) {
            xp[(long)(m0 + rbase + r) * HID + n] = c[r] + bias;
        }
    }
}

// ---------------------------------------------------------------------------
// Phase 2: sequential scan, single 256-thread block (8 waves -> 2 waves/SIMD,
// so each thread can pin its whole wh_w row (64 x v4f = 256 VGPRs) in registers).
// Per step: 64 LDS-broadcast b128 loads of h + 256 register FMAs (4 rotating
// accumulators), sigmoid, one barrier. Zero global traffic on the recurrence
// except the prefetched xp[t+1].
// ---------------------------------------------------------------------------
__global__ __launch_bounds__(256) void srn_scan_kernel(
    const float* __restrict__ xp,
    const float* __restrict__ wh_w,
    const float* __restrict__ wh_b,
    float* __restrict__ out)
{
    __shared__ float hbuf[2][HID];

    const int j = threadIdx.x;
    const float bias = wh_b[j];
    const float* wrow = wh_w + (long)j * HID;

    // Pin the full W row in VGPRs (fully unrolled constant indexing -> no spills).
    v4f w[64];
    #pragma unroll
    for (int i = 0; i < 64; ++i) w[i] = *(const v4f*)(wrow + 4 * i);

    hbuf[0][j] = 0.0f;          // h0 = zeros
    __syncthreads();

    int cur = 0;
    float xcur = xp[j];         // xp[0][j]
    for (int tstep = 0; tstep < SEQ; ++tstep) {
        // issue next step's xp load early to hide its latency behind the matvec
        float xnxt = 0.0f;
        if (tstep + 1 < SEQ) xnxt = xp[(size_t)(tstep + 1) * HID + j];

        float accs[4] = {xcur + bias, 0.0f, 0.0f, 0.0f};
        #pragma unroll
        for (int i = 0; i < 64; ++i) {
            v4f h4 = *(const v4f*)(&hbuf[cur][4 * i]);   // LDS broadcast (same addr all lanes)
            float s = accs[i & 3];
            s = __builtin_fmaf(h4.x, w[i].x, s);
            s = __builtin_fmaf(h4.y, w[i].y, s);
            s = __builtin_fmaf(h4.z, w[i].z, s);
            s = __builtin_fmaf(h4.w, w[i].w, s);
            accs[i & 3] = s;
        }
        const float acc = (accs[0] + accs[1]) + (accs[2] + accs[3]);

        const float hn = 1.0f / (1.0f + __expf(-acc));   // sigmoid via v_exp_f32
        hbuf[cur ^ 1][j] = hn;
        __syncthreads();        // writes to nxt visible; reads of cur all done before here
        cur ^= 1;
        xcur = xnxt;
    }

    out[j] = hbuf[cur][j];
}

// ---------------------------------------------------------------------------
extern "C" void kernel_launch(void* const* d_in, const int* in_sizes, int n_in,
                              void* d_out, int out_size, void* d_ws, size_t ws_size,
                              hipStream_t stream) {
    const int*   tokens    = (const int*)  d_in[0];   // (SEQ,) int32
    const float* embedding = (const float*)d_in[1];   // (VOCAB, EMB) f32
    const float* wx_w      = (const float*)d_in[2];   // (HID, EMB) f32
    const float* wx_b      = (const float*)d_in[3];   // (HID,) f32
    const float* wh_w      = (const float*)d_in[4];   // (HID, HID) f32
    const float* wh_b      = (const float*)d_in[5];   // (HID,) f32
    float*       out       = (float*)d_out;           // (1, HID) f32
    float*       xp        = (float*)d_ws;            // scratch: SEQ*HID f32 = 4 MB

    xproj_wmma_kernel<<<SEQ / 16, 256, 0, stream>>>(tokens, embedding, wx_w, wx_b, xp);
    srn_scan_kernel<<<1, 256, 0, stream>>>(xp, wh_w, wh_b, out);
}